// GCN_90099823935876
// MI455X (gfx1250) — compile-verified
//
#include <hip/hip_runtime.h>

#define N_NODES 50000
#define N_EDGES 800000
#define IN_C    128
#define HID_C   96   // == OUT_C

typedef __attribute__((ext_vector_type(2))) float v2f;
typedef __attribute__((ext_vector_type(8))) float v8f;

// ---------------- degree / normalization ----------------

__global__ void k_zero(float* __restrict__ p, int n) {
  int i = blockIdx.x * blockDim.x + threadIdx.x;
  if (i < n) p[i] = 0.0f;
}

__global__ void k_degree(const int* __restrict__ dst, float* __restrict__ deg, int E) {
  int e = blockIdx.x * blockDim.x + threadIdx.x;
  if (e < E) atomicAdd(&deg[dst[e]], 1.0f);
}

__global__ void k_rsqrt(float* __restrict__ deg, int n) {
  int i = blockIdx.x * blockDim.x + threadIdx.x;
  if (i < n) deg[i] = rsqrtf(deg[i] + 1.0f);  // +1 for self-loop
}

// ---------------- fp32 WMMA GEMM: H[M x 96] = A[M x K] * B[K x 96] ----------------
// One wave per 16-row strip of A; 6 column tiles (N=96) accumulated in registers.
// Uses V_WMMA_F32_16X16X4_F32 (exact fp32 matrix op, CDNA5).
__global__ void k_gemm_wmma(const float* __restrict__ A, const float* __restrict__ B,
                            float* __restrict__ H, int M, int K) {
  const int NW = 96;
  int wave = (int)((blockIdx.x * blockDim.x + threadIdx.x) >> 5);
  int lane = threadIdx.x & 31;
  int nRowTiles = M >> 4;
  if (wave >= nRowTiles) return;

  int row0 = wave << 4;
  int half = lane >> 4;   // 0: K pair {k,k+1}; 1: K pair {k+2,k+3}
  int l    = lane & 15;   // M index for A, N index within tile for B/C/D

  v8f acc[6] = {};

  // A row pointer for this lane (even-float offset -> 8B aligned pairs)
  const float* __restrict__ arow = A + (size_t)(row0 + l) * K + half * 2;

  for (int k = 0; k < K; k += 4) {
    v2f a;
    a.x = arow[k];
    a.y = arow[k + 1];
    const float* __restrict__ b0 = B + (size_t)(k + half * 2) * NW + l;
#pragma unroll
    for (int t = 0; t < 6; ++t) {
      v2f b;
      b.x = b0[t * 16];        // row (k + half*2),     col t*16 + l
      b.y = b0[t * 16 + NW];   // row (k + half*2 + 1), col t*16 + l
      // 8 args: (neg_a, A, neg_b, B, c_mod, C, reuse_a, reuse_b)
      acc[t] = __builtin_amdgcn_wmma_f32_16x16x4_f32(
          false, a, false, b, (short)0, acc[t], false, false);
    }
  }

  // C/D layout: VGPR r -> M = r + 8*half, N = l (within tile)
#pragma unroll
  for (int t = 0; t < 6; ++t) {
    float* __restrict__ hp = H + (size_t)(row0 + half * 8) * NW + t * 16 + l;
#pragma unroll
    for (int r = 0; r < 8; ++r)
      hp[(size_t)r * NW] = acc[t][r];
  }
}

// ---------------- self-loop + bias init: agg = h * dinv^2 + b ----------------
__global__ void k_init_agg(const float* __restrict__ h, const float* __restrict__ dinv,
                           const float* __restrict__ bias, float* __restrict__ agg,
                           int total /* = N_NODES*96 */) {
  int i = blockIdx.x * blockDim.x + threadIdx.x;
  if (i >= total) return;
  int node = i / 96;
  int c = i - node * 96;
  float di = dinv[node];
  agg[i] = fmaf(h[i], di * di, bias[c]);
}

// ---------------- edge scatter: agg[dst] += h[src] * dinv[src]*dinv[dst] ----------------
// One wave per edge; 96 channels = 3 coalesced f32 atomic adds per lane (wave32).
__global__ void k_edge_scatter(const float* __restrict__ h, const int* __restrict__ src,
                               const int* __restrict__ dst, const float* __restrict__ dinv,
                               float* __restrict__ agg, int E) {
  int w = (int)((blockIdx.x * blockDim.x + threadIdx.x) >> 5);
  int lane = threadIdx.x & 31;
  if (w >= E) return;
  int s = src[w];
  int d = dst[w];
  float norm = dinv[s] * dinv[d];
  const float* __restrict__ hs = h + (size_t)s * 96;
  float* __restrict__ ad = agg + (size_t)d * 96;
#pragma unroll
  for (int c = 0; c < 3; ++c) {
    int ch = lane + 32 * c;
    atomicAdd(&ad[ch], hs[ch] * norm);
  }
}

__global__ void k_relu(float* __restrict__ p, int n) {
  int i = blockIdx.x * blockDim.x + threadIdx.x;
  if (i < n) p[i] = fmaxf(p[i], 0.0f);
}

// ---------------- driver ----------------

extern "C" void kernel_launch(void* const* d_in, const int* in_sizes, int n_in,
                              void* d_out, int out_size, void* d_ws, size_t ws_size,
                              hipStream_t stream) {
  (void)in_sizes; (void)n_in; (void)out_size; (void)ws_size;

  const float* x   = (const float*)d_in[0];
  const int*   ei  = (const int*)d_in[1];   // (2, E) row-major int32
  const float* W1  = (const float*)d_in[2];
  const float* b1  = (const float*)d_in[3];
  const float* W2  = (const float*)d_in[4];
  const float* b2  = (const float*)d_in[5];
  const float* W3  = (const float*)d_in[6];
  const float* b3  = (const float*)d_in[7];
  const float* W4  = (const float*)d_in[8];
  const float* b4  = (const float*)d_in[9];
  float* out = (float*)d_out;

  const int* src = ei;
  const int* dst = ei + N_EDGES;

  // Workspace: dinv [N] | h [N*96] | aggA [N*96]   (~38.6 MB)
  float* dinv = (float*)d_ws;
  float* h    = dinv + N_NODES;
  float* aggA = h + (size_t)N_NODES * HID_C;

  const int elems = N_NODES * HID_C;               // 4.8M
  const int ebElems = (elems + 255) / 256;
  const int ebNodes = (N_NODES + 255) / 256;
  const int ebEdges = (N_EDGES + 255) / 256;
  const int gemmBlocks = ((N_NODES / 16) * 32 + 255) / 256;  // 3125 waves
  const int scatterBlocks = (N_EDGES * 32 + 255) / 256;      // 1 wave per edge

  // degree -> dinv (once; shared by all layers)
  k_zero<<<ebNodes, 256, 0, stream>>>(dinv, N_NODES);
  k_degree<<<ebEdges, 256, 0, stream>>>(dst, dinv, N_EDGES);
  k_rsqrt<<<ebNodes, 256, 0, stream>>>(dinv, N_NODES);

  // Layer 1: x(128) -> aggA
  k_gemm_wmma<<<gemmBlocks, 256, 0, stream>>>(x, W1, h, N_NODES, IN_C);
  k_init_agg<<<ebElems, 256, 0, stream>>>(h, dinv, b1, aggA, elems);
  k_edge_scatter<<<scatterBlocks, 256, 0, stream>>>(h, src, dst, dinv, aggA, N_EDGES);
  k_relu<<<ebElems, 256, 0, stream>>>(aggA, elems);

  // Layer 2: aggA -> out (used as scratch; overwritten by layer 4)
  k_gemm_wmma<<<gemmBlocks, 256, 0, stream>>>(aggA, W2, h, N_NODES, HID_C);
  k_init_agg<<<ebElems, 256, 0, stream>>>(h, dinv, b2, out, elems);
  k_edge_scatter<<<scatterBlocks, 256, 0, stream>>>(h, src, dst, dinv, out, N_EDGES);
  k_relu<<<ebElems, 256, 0, stream>>>(out, elems);

  // Layer 3: out -> aggA
  k_gemm_wmma<<<gemmBlocks, 256, 0, stream>>>(out, W3, h, N_NODES, HID_C);
  k_init_agg<<<ebElems, 256, 0, stream>>>(h, dinv, b3, aggA, elems);
  k_edge_scatter<<<scatterBlocks, 256, 0, stream>>>(h, src, dst, dinv, aggA, N_EDGES);
  k_relu<<<ebElems, 256, 0, stream>>>(aggA, elems);

  // Layer 4: aggA -> out (no ReLU)
  k_gemm_wmma<<<gemmBlocks, 256, 0, stream>>>(aggA, W4, h, N_NODES, HID_C);
  k_init_agg<<<ebElems, 256, 0, stream>>>(h, dinv, b4, out, elems);
  k_edge_scatter<<<scatterBlocks, 256, 0, stream>>>(h, src, dst, dinv, out, N_EDGES);
}